// SAGE_61022895342087
// MI455X (gfx1250) — compile-verified
//
#include <hip/hip_runtime.h>
#include <hip/hip_bf16.h>

typedef __attribute__((ext_vector_type(2))) float v2f;
typedef __attribute__((ext_vector_type(8))) float v8f;

#define IN_SIZE 128
#define HID     256
#define ND1     100000
#define E1      1600000
#define ND2     16384
#define E2      262144

// -------------------- zero fill (float4 grid-stride) --------------------
__global__ __launch_bounds__(256) void SAGE_zero(float4* __restrict__ p, long n4) {
    long i = (long)blockIdx.x * blockDim.x + threadIdx.x;
    long stride = (long)gridDim.x * blockDim.x;
    float4 z = {0.f, 0.f, 0.f, 0.f};
    for (; i < n4; i += stride) p[i] = z;
}

// -------------------- scatter-add (mean numerator + degree) -------------
// One wave32 per edge; each lane moves D/32 floats via float4 loads and scalar float atomics.
template<int D>
__global__ __launch_bounds__(256) void SAGE_scatter(
    const float* __restrict__ feat, const int* __restrict__ src,
    const int* __restrict__ dst, float* __restrict__ msg,
    float* __restrict__ deg, int nE)
{
    int gid  = blockIdx.x * blockDim.x + threadIdx.x;
    int e    = gid >> 5;
    int lane = gid & 31;
    if (e >= nE) return;
    int s = src[e];
    int d = dst[e];
    const int NV4 = D / 128;  // float4s per lane (1 for D=128, 2 for D=256)
    const float4* fp = (const float4*)(feat + (size_t)s * D);
    float*        mp = msg + (size_t)d * D;
#pragma unroll
    for (int c = 0; c < NV4; ++c) {
        int v4 = lane * NV4 + c;
        float4 v = fp[v4];
        float* m = mp + v4 * 4;
        atomicAdd(m + 0, v.x);
        atomicAdd(m + 1, v.y);
        atomicAdd(m + 2, v.z);
        atomicAdd(m + 3, v.w);
    }
    if (lane == 0) atomicAdd(deg + d, 1.0f);
}

// -------------------- fused SAGE GEMM via V_WMMA_F32_16X16X4_F32 --------
// out[m, 0:256] = act( Aself[m,:K] @ Wself + (Amsg[m,:K]/max(deg,1)) @ Wneigh + bias )
// Block: 128 threads = 4 waves. Block owns 16 M-rows; wave w owns N in
// [64w, 64w+64) as four 16x16 WMMA tiles. Both matmuls accumulate into the
// same C registers (sum of two GEMMs).
template<int K, bool RELU>
__global__ __launch_bounds__(128) void SAGE_gemm(
    const float* __restrict__ Aself, const float* __restrict__ Amsg,
    const float* __restrict__ deg,   const float* __restrict__ Wself,
    const float* __restrict__ Wneigh,const float* __restrict__ bias,
    float* __restrict__ out)
{
    const int N = 256;
    const int m0   = blockIdx.x * 16;
    const int wave = threadIdx.x >> 5;
    const int lane = threadIdx.x & 31;
    const int r    = lane & 15;   // M index for A, N-offset for B/C
    const int half = lane >> 4;   // 0: K={k,k+1}/M=g ; 1: K={k+2,k+3}/M=g+8
    const int row  = m0 + r;
    const int n0w  = wave * 64;

    float dv = deg[row];
    float scl = 1.0f / fmaxf(dv, 1.0f);

    v8f acc[4];
#pragma unroll
    for (int t = 0; t < 4; ++t)
#pragma unroll
        for (int g = 0; g < 8; ++g) acc[t][g] = 0.0f;

    const float* aS = Aself + (size_t)row * K;
    const float* aM = Amsg  + (size_t)row * K;

    for (int k = 0; k < K; k += 4) {
        const int ka = k + 2 * half;
        // A operands: 16x4 f32, lane<16 -> K={k,k+1}, lane>=16 -> K={k+2,k+3}
        v2f a_s = *(const v2f*)(aS + ka);
        v2f am  = *(const v2f*)(aM + ka);
        v2f a_n;
        a_n.x = am.x * scl;
        a_n.y = am.y * scl;
#pragma unroll
        for (int t = 0; t < 4; ++t) {
            const int n = n0w + t * 16 + r;
            // B operands: 4x16 f32; VGPR0 = rows k / k+2, VGPR1 = rows k+1 / k+3
            const float* ws = Wself  + (size_t)ka * N + n;
            const float* wn = Wneigh + (size_t)ka * N + n;
            v2f b_s; b_s.x = ws[0]; b_s.y = ws[N];
            v2f b_n; b_n.x = wn[0]; b_n.y = wn[N];
            acc[t] = __builtin_amdgcn_wmma_f32_16x16x4_f32(
                false, a_s, false, b_s, (short)0, acc[t], false, false);
            acc[t] = __builtin_amdgcn_wmma_f32_16x16x4_f32(
                false, a_n, false, b_n, (short)0, acc[t], false, false);
        }
    }

    // C/D layout: VGPR g, lanes 0-15 -> M=g, lanes 16-31 -> M=g+8, N = n0+lane%16
#pragma unroll
    for (int t = 0; t < 4; ++t) {
        const int n = n0w + t * 16 + r;
        float bv = bias[n];
#pragma unroll
        for (int g = 0; g < 8; ++g) {
            float v = acc[t][g] + bv;
            if (RELU) v = fmaxf(v, 0.0f);
            out[(size_t)(m0 + g + 8 * half) * N + n] = v;
        }
    }
}

extern "C" void kernel_launch(void* const* d_in, const int* in_sizes, int n_in,
                              void* d_out, int out_size, void* d_ws, size_t ws_size,
                              hipStream_t stream) {
    const float* x        = (const float*)d_in[0];
    const int*   src1     = (const int*)d_in[1];
    const int*   dst1     = (const int*)d_in[2];
    const int*   src2     = (const int*)d_in[3];
    const int*   dst2     = (const int*)d_in[4];
    const float* W_self1  = (const float*)d_in[5];
    const float* W_neigh1 = (const float*)d_in[6];
    const float* b1       = (const float*)d_in[7];
    const float* W_self2  = (const float*)d_in[8];
    const float* W_neigh2 = (const float*)d_in[9];
    const float* b2       = (const float*)d_in[10];
    float* out = (float*)d_out;

    // Workspace layout (floats)
    float* ws   = (float*)d_ws;
    float* msg1 = ws;                                   // ND1*128 = 12,800,000
    float* deg1 = msg1 + (size_t)ND1 * IN_SIZE;         // ND1     =    100,000
    float* h1   = deg1 + ND1;                           // ND1*256 = 25,600,000
    float* msg2 = h1 + (size_t)ND1 * HID;               // ND2*256 =  4,194,304
    float* deg2 = msg2 + (size_t)ND2 * HID;             // ND2     =     16,384

    // 1) zero accumulators (msg1+deg1 contiguous; msg2+deg2 contiguous)
    {
        long n4a = ((long)ND1 * IN_SIZE + ND1) / 4;
        long n4b = ((long)ND2 * HID + ND2) / 4;
        SAGE_zero<<<2048, 256, 0, stream>>>((float4*)msg1, n4a);
        SAGE_zero<<<1024, 256, 0, stream>>>((float4*)msg2, n4b);
    }

    // 2) layer-1 scatter: one wave per edge
    {
        long threads = (long)E1 * 32;
        int blocks = (int)((threads + 255) / 256);
        SAGE_scatter<IN_SIZE><<<blocks, 256, 0, stream>>>(x, src1, dst1, msg1, deg1, E1);
    }

    // 3) layer-1 fused GEMM + bias + relu -> h1
    SAGE_gemm<IN_SIZE, true><<<ND1 / 16, 128, 0, stream>>>(
        x, msg1, deg1, W_self1, W_neigh1, b1, h1);

    // 4) layer-2 scatter
    {
        long threads = (long)E2 * 32;
        int blocks = (int)((threads + 255) / 256);
        SAGE_scatter<HID><<<blocks, 256, 0, stream>>>(h1, src2, dst2, msg2, deg2, E2);
    }

    // 5) layer-2 fused GEMM + bias -> out
    SAGE_gemm<HID, false><<<ND2 / 16, 128, 0, stream>>>(
        h1, msg2, deg2, W_self2, W_neigh2, b2, out);
}